// QuanvLayer1D_39642548142544
// MI455X (gfx1250) — compile-verified
//
#include <hip/hip_runtime.h>
#include <hip/hip_bf16.h>

typedef __attribute__((ext_vector_type(2))) float v2f;
typedef __attribute__((ext_vector_type(8))) float v8f;

struct c32 { float x, y; };
__device__ __forceinline__ c32 cmul(c32 a, c32 b) { return { a.x * b.x - a.y * b.y, a.x * b.y + a.y * b.x }; }
__device__ __forceinline__ c32 cadd(c32 a, c32 b) { return { a.x + b.x, a.y + b.y }; }

// -------- wave-parallel 8-qubit statevector sim (256 amps: 8 per lane, wave32) --------
// flat basis index b in [0,256): bit for qubit j is (b >> (7-j)) & 1.
// lane holds b = lane*8 + r, r in [0,8): bits 2..0 of b are register bits, bits 7..3 lane bits.

__device__ __forceinline__ void gate1q(c32* a, int m, c32 u00, c32 u01, c32 u10, c32 u11,
                                       unsigned lane) {
  if (m < 3) {                       // register-bit qubit: in-lane 2x2
    const int tm = 1 << m;
#pragma unroll
    for (int r = 0; r < 8; ++r) {
      if (!(r & tm)) {
        c32 a0 = a[r], a1 = a[r | tm];
        a[r]      = cadd(cmul(u00, a0), cmul(u01, a1));
        a[r | tm] = cadd(cmul(u10, a0), cmul(u11, a1));
      }
    }
  } else {                           // lane-bit qubit: cross-lane via shfl_xor
    const int lm = 1 << (m - 3);
    const bool hi = (lane & (unsigned)lm) != 0u;
#pragma unroll
    for (int r = 0; r < 8; ++r) {
      c32 me = a[r], p;
      p.x = __shfl_xor(me.x, lm, 32);
      p.y = __shfl_xor(me.y, lm, 32);
      a[r] = hi ? cadd(cmul(u10, p), cmul(u11, me))
                : cadd(cmul(u00, me), cmul(u01, p));
    }
  }
}

__device__ __forceinline__ void gcnot(c32* a, int cb, int tb, unsigned lane) {
  c32 old[8];
#pragma unroll
  for (int r = 0; r < 8; ++r) old[r] = a[r];
#pragma unroll
  for (int r = 0; r < 8; ++r) {
    unsigned b = lane * 8u + (unsigned)r;
    c32 part;
    if (tb < 3) {
      part = old[r ^ (1 << tb)];
    } else {
      const int lm = 1 << (tb - 3);
      part.x = __shfl_xor(old[r].x, lm, 32);
      part.y = __shfl_xor(old[r].y, lm, 32);
    }
    a[r] = ((b >> cb) & 1u) ? part : old[r];
  }
}

__device__ __forceinline__ void run_ansatz(c32* a, const float* __restrict__ W, unsigned lane) {
#pragma unroll
  for (int j = 0; j < 8; ++j) {
    const int m  = 7 - j;                       // qubit j bit position
    const int cb = (j < 7) ? 7 - j : 7;         // control: qubit j (or 0 for j==7)
    const int tb = (j < 7) ? 6 - j : 0;         // target:  qubit j+1 (or 7 for j==7)
    float s, c;
    sincosf(0.5f * W[j * 3 + 0], &s, &c);       // RX
    gate1q(a, m, c32{c, 0.f}, c32{0.f, -s}, c32{0.f, -s}, c32{c, 0.f}, lane);
    gcnot(a, cb, tb, lane);
    sincosf(0.5f * W[j * 3 + 1], &s, &c);       // RY
    gate1q(a, m, c32{c, 0.f}, c32{-s, 0.f}, c32{s, 0.f}, c32{c, 0.f}, lane);
    gcnot(a, cb, tb, lane);
    sincosf(0.5f * W[j * 3 + 2], &s, &c);       // RZ = diag(e^{-it/2}, e^{+it/2})
    gate1q(a, m, c32{c, -s}, c32{0.f, 0.f}, c32{0.f, 0.f}, c32{c, s}, lane);
  }
}

// ws layout (floats): [0,512) v0 interleaved (re,im);
// then per matrix m=0..3: Ar[256*256] followed by Ai[256*256]  (row-major, row=out basis, col=in basis)
__global__ __launch_bounds__(128) void qc_precompute(const float* __restrict__ weights,
                                                     float* __restrict__ ws) {
  unsigned gt = blockIdx.x * blockDim.x + threadIdx.x;
  unsigned wv = gt >> 5, lane = gt & 31u;
  if (wv >= 1025u) return;                      // wave-uniform exit
  unsigned s_col = (wv == 0u) ? 0u : ((wv - 1u) & 255u);
  unsigned mat   = (wv == 0u) ? 0u : ((wv - 1u) >> 8);   // 0..3 -> weights layer mat+1
  c32 a[8];
#pragma unroll
  for (int r = 0; r < 8; ++r) {
    unsigned b = lane * 8u + (unsigned)r;
    a[r].x = (b == s_col) ? 1.0f : 0.0f;
    a[r].y = 0.0f;
  }
  const float* W = weights + ((wv == 0u) ? 0u : (mat + 1u)) * 24u;
  run_ansatz(a, W, lane);
  if (wv == 0u) {
#pragma unroll
    for (int r = 0; r < 8; ++r) {
      unsigned b = lane * 8u + (unsigned)r;
      ws[2u * b]      = a[r].x;
      ws[2u * b + 1u] = a[r].y;
    }
  } else {
    float* Ar = ws + 512u + mat * 131072u;
    float* Ai = Ar + 65536u;
#pragma unroll
    for (int r = 0; r < 8; ++r) {
      unsigned b = lane * 8u + (unsigned)r;
      Ar[b * 256u + s_col] = a[r].x;            // column s_col of the operator
      Ai[b * 256u + s_col] = a[r].y;
    }
  }
}

__global__ void qc_zero(float* __restrict__ out) { out[threadIdx.x] = 0.0f; }

// 16 patches per workgroup; 16 waves: wave p owns patch-column p AND output row-tile p.
// Psi stored column-major in LDS: S[col*256 + row].
__global__ __launch_bounds__(512) void qc_main(const float* __restrict__ vec,
                                               const float* __restrict__ ws,
                                               float* __restrict__ out) {
  __shared__ float Sr[16 * 256];
  __shared__ float Si[16 * 256];
  const unsigned tid  = threadIdx.x;
  const unsigned wv   = tid >> 5;               // 0..15
  const unsigned lane = tid & 31u;
  const unsigned pg   = blockIdx.x * 16u + wv;  // global patch id, < 4096
  const bool active   = pg < 4088u;             // 8 batches * 511 positions
  const unsigned batch = pg / 511u;
  const unsigned pos   = pg % 511u;

  // patch angles (PAD=1 zero padding), precompute sin/cos of x/2 for the RY uploads
  float gc[4], gs[4];
#pragma unroll
  for (int k = 0; k < 4; ++k) {
    int ii = (int)pos + k - 1;
    bool inb = active && (ii >= 0) && (ii < 512);
    unsigned ld = inb ? (batch * 512u + (unsigned)ii) : 0u;
    float xv = vec[ld];
    xv = inb ? xv : 0.0f;
    sincosf(0.5f * xv, &gs[k], &gc[k]);
  }

  // init column wv with v0 (zeros if inactive)
  float* myR = Sr + wv * 256u;
  float* myI = Si + wv * 256u;
#pragma unroll
  for (int r = 0; r < 8; ++r) {
    unsigned b = lane * 8u + (unsigned)r;
    float vr = ws[2u * b], vi = ws[2u * b + 1u];
    myR[b] = active ? vr : 0.0f;
    myI[b] = active ? vi : 0.0f;
  }
  __syncthreads();

  const unsigned l15 = lane & 15u;
  const unsigned kb  = (lane >> 4) * 2u;        // WMMA A/B frag K sub-offset per half-wave

  for (int layer = 0; layer < 4; ++layer) {
    // ---- data re-upload D(x): RY(x[q&3]) on qubit q (bit 7-q), own column only ----
#pragma unroll
    for (int q = 0; q < 8; ++q) {
      const int m = 7 - q;
      const float c = gc[q & 3], s = gs[q & 3];
#pragma unroll
      for (int t = 0; t < 4; ++t) {
        unsigned pi  = lane * 4u + (unsigned)t;           // pair index 0..127
        unsigned low = pi & ((1u << m) - 1u);
        unsigned b0  = ((pi ^ low) << 1) | low;           // insert 0 at bit m
        unsigned b1  = b0 | (1u << m);
        float a0r = myR[b0], a1r = myR[b1];
        float a0i = myI[b0], a1i = myI[b1];
        myR[b0] = c * a0r - s * a1r;
        myR[b1] = s * a0r + c * a1r;
        myI[b0] = c * a0i - s * a1i;
        myI[b1] = s * a0i + c * a1i;
      }
    }
    __syncthreads();

    // ---- Psi = A_layer * Psi : complex (256x256)*(256x16) via V_WMMA_F32_16X16X4_F32 ----
    const float* Ar = ws + 512u + (unsigned)layer * 131072u;
    const float* Ai = Ar + 65536u;
    const unsigned arow = (wv * 16u + l15) * 256u;        // A row for this lane
    v8f acc_rr = {}, acc_ii = {}, acc_ri = {}, acc_ir = {};
#pragma unroll 2
    for (unsigned kk = 0u; kk < 256u; kk += 4u) {
      v2f afr = *(const v2f*)(Ar + arow + kk + kb);       // A[M=l15][K=kk+kb..+1]
      v2f afi = *(const v2f*)(Ai + arow + kk + kb);
      v2f bfr = *(const v2f*)(Sr + l15 * 256u + kk + kb); // B[K][N=l15] (col-major LDS)
      v2f bfi = *(const v2f*)(Si + l15 * 256u + kk + kb);
      acc_rr = __builtin_amdgcn_wmma_f32_16x16x4_f32(false, afr, false, bfr, (short)0, acc_rr, false, false);
      acc_ii = __builtin_amdgcn_wmma_f32_16x16x4_f32(false, afi, false, bfi, (short)0, acc_ii, false, false);
      acc_ri = __builtin_amdgcn_wmma_f32_16x16x4_f32(false, afr, false, bfi, (short)0, acc_ri, false, false);
      acc_ir = __builtin_amdgcn_wmma_f32_16x16x4_f32(false, afi, false, bfr, (short)0, acc_ir, false, false);
    }
    v8f cr = acc_rr - acc_ii;                             // Re = Ar*Pr - Ai*Pi
    v8f ci = acc_ri + acc_ir;                             // Im = Ar*Pi + Ai*Pr
    __syncthreads();                                      // all reads of Psi done
    // in-place writeback: C element (M = v + 8*(lane>>4), N = l15), rows wv*16..+15
    const unsigned rbase = wv * 16u + (lane >> 4) * 8u;
#pragma unroll
    for (int v = 0; v < 8; ++v) {
      Sr[l15 * 256u + rbase + (unsigned)v] = cr[v];
      Si[l15 * 256u + rbase + (unsigned)v] = ci[v];
    }
    __syncthreads();
  }

  // ---- measurement: z_j = sum_b |psi_b|^2 * (1 - 2*bit_j(b)); relu; max over positions ----
  float z[8];
#pragma unroll
  for (int j = 0; j < 8; ++j) z[j] = 0.0f;
#pragma unroll
  for (int r = 0; r < 8; ++r) {
    unsigned b = lane * 8u + (unsigned)r;
    float re = myR[b], im = myI[b];
    float pr = re * re + im * im;
#pragma unroll
    for (int j = 0; j < 8; ++j)
      z[j] += ((b >> (7 - j)) & 1u) ? -pr : pr;
  }
#pragma unroll
  for (int off = 16; off > 0; off >>= 1) {
#pragma unroll
    for (int j = 0; j < 8; ++j) z[j] += __shfl_xor(z[j], off, 32);
  }
  if (lane == 0u && active) {
#pragma unroll
    for (int j = 0; j < 8; ++j) {
      unsigned bits = __float_as_uint(fmaxf(z[j], 0.0f));   // >=0: bit pattern monotone
      atomicMax((unsigned*)(out + batch * 8u + (unsigned)j), bits);
    }
  }
}

extern "C" void kernel_launch(void* const* d_in, const int* in_sizes, int n_in,
                              void* d_out, int out_size, void* d_ws, size_t ws_size,
                              hipStream_t stream) {
  (void)in_sizes; (void)n_in; (void)out_size; (void)ws_size;
  const float* vec     = (const float*)d_in[0];   // (8,1,512) f32
  const float* weights = (const float*)d_in[1];   // (5,8,3)  f32
  float* out = (float*)d_out;                     // (8,8,1)  f32
  float* ws  = (float*)d_ws;                      // needs 512 + 4*2*65536 floats (~2.1 MB)

  qc_zero<<<1, 64, 0, stream>>>(out);
  qc_precompute<<<257, 128, 0, stream>>>(weights, ws);    // 1025 waves: v0 + 4 dense 256x256 ops
  qc_main<<<256, 512, 0, stream>>>(vec, ws, out);         // 16 patches/block, WMMA GEMM core
}